// CapsuleLayer_25855703122272
// MI455X (gfx1250) — compile-verified
//
#include <hip/hip_runtime.h>
#include <math.h>

// Problem constants (from reference)
#define BATCH   32
#define NIN     2048
#define KDIM    16
#define NCAP    32
#define CAPD    32
#define NOUT    1024          // NCAP * CAPD
#define NCHUNK  128           // workgroups along n
#define NC      (NIN / NCHUNK) // 16 n per workgroup
#define BH      16            // batches per workgroup (one WMMA M tile)
#define UHS     1028          // LDS row stride for u_hat (bank-conflict pad)

typedef float v2f __attribute__((ext_vector_type(2)));
typedef float v8f __attribute__((ext_vector_type(8)));

// One pass of routing. Each WG: 16 batches (blockIdx.y half) x 16 n's (blockIdx.x chunk),
// all 1024 output columns. Recomputes u_hat with V_WMMA_F32_16X16X4_F32, keeps it in LDS,
// does logits -> softmax -> weighted accumulation, writes per-WG partial o to workspace.
__global__ __launch_bounds__(256)
void caps_pass(const float* __restrict__ U,      // [32][2048][16]
               const float* __restrict__ Wt,     // [2048][16][1024]
               const float* __restrict__ o_prev, // [32][32][32] (unused when iter0)
               float* __restrict__ partial,      // [256][16*1024]
               int iter0)
{
    extern __shared__ float smem[];
    float* uhat = smem;                    // [BH][UHS]
    float* oacc = uhat + BH * UHS;         // [BH*1024]
    float* lg   = oacc + BH * 1024;        // [BH*32]
    float* cb   = lg + BH * 32;            // [BH*32]

    const int tid   = threadIdx.x;
    const int lane  = tid & 31;
    const int wave  = tid >> 5;            // 0..7, owns cols [wave*128, wave*128+128)
    const int chunk = blockIdx.x;          // 0..127
    const int half  = blockIdx.y;          // 0..1
    const int b0    = half * BH;

    const int l16 = lane & 15;
    const int hi  = lane >> 4;             // half-wave select (K offset / row offset)

    for (int e = tid; e < BH * 1024; e += 256) oacc[e] = 0.0f;
    __syncthreads();

    for (int nn = 0; nn < NC; ++nn) {
        const int n = chunk * NC + nn;

        if (nn + 1 < NC) {  // prefetch next n's W slice for this wave's columns
            __builtin_prefetch(Wt + ((size_t)(n + 1) * KDIM) * NOUT + wave * 128, 0, 3);
        }

        // ---- A fragments (16x4 tiles over K=16): lane row = l16, K base 2*hi ----
        const float* Up = U + ((size_t)(b0 + l16) * NIN + n) * KDIM;
        v2f a0, a1, a2, a3;
        a0.x = Up[0  + 2*hi]; a0.y = Up[1  + 2*hi];
        a1.x = Up[4  + 2*hi]; a1.y = Up[5  + 2*hi];
        a2.x = Up[8  + 2*hi]; a2.y = Up[9  + 2*hi];
        a3.x = Up[12 + 2*hi]; a3.y = Up[13 + 2*hi];

        const float* Wn = Wt + (size_t)n * KDIM * NOUT;
        for (int ct = 0; ct < 8; ++ct) {
            const int c0 = wave * 128 + ct * 16;
            // B fragments (4x16): VGPR0 rows {0,2}+4s, VGPR1 rows {1,3}+4s
            const float* Wp = Wn + (size_t)(2 * hi) * NOUT + c0 + l16;
            v2f bf0, bf1, bf2, bf3;
            bf0.x = Wp[0  * NOUT]; bf0.y = Wp[1  * NOUT];
            bf1.x = Wp[4  * NOUT]; bf1.y = Wp[5  * NOUT];
            bf2.x = Wp[8  * NOUT]; bf2.y = Wp[9  * NOUT];
            bf3.x = Wp[12 * NOUT]; bf3.y = Wp[13 * NOUT];

            v8f acc = {};
            acc = __builtin_amdgcn_wmma_f32_16x16x4_f32(false, a0, false, bf0, (short)0, acc, false, false);
            acc = __builtin_amdgcn_wmma_f32_16x16x4_f32(false, a1, false, bf1, (short)0, acc, false, false);
            acc = __builtin_amdgcn_wmma_f32_16x16x4_f32(false, a2, false, bf2, (short)0, acc, false, false);
            acc = __builtin_amdgcn_wmma_f32_16x16x4_f32(false, a3, false, bf3, (short)0, acc, false, false);

            // C/D layout: VGPR v -> local batch row v + 8*hi, col c0 + l16
            #pragma unroll
            for (int v = 0; v < 8; ++v)
                uhat[(v + 8 * hi) * UHS + c0 + l16] = acc[v];
        }
        __syncthreads();

        if (iter0) {
            // round 0: softmax(0) == 1/32 uniform coupling
            for (int e = tid; e < BH * 1024; e += 256)
                oacc[e] += 0.03125f * uhat[(e >> 10) * UHS + (e & 1023)];
        } else {
            // logits b[bl,i] = sum_j o_prev[b,i,j] * u_hat[b,n,i,j]
            for (int p = tid; p < BH * NCAP; p += 256) {
                const int bl = p >> 5, i = p & 31;
                const float* op = o_prev + ((size_t)(b0 + bl) * NCAP + i) * CAPD;
                const float* uh = uhat + bl * UHS + i * CAPD;
                float s = 0.0f;
                #pragma unroll
                for (int j = 0; j < CAPD; ++j) s = fmaf(op[j], uh[j], s);
                lg[p] = s;
            }
            __syncthreads();
            // softmax over capsules i per local batch (16 rows)
            if (tid < BH) {
                float m = -3.402823466e38f;
                for (int i = 0; i < NCAP; ++i) m = fmaxf(m, lg[tid * NCAP + i]);
                float s = 0.0f;
                for (int i = 0; i < NCAP; ++i) {
                    float e = __expf(lg[tid * NCAP + i] - m);
                    cb[tid * NCAP + i] = e; s += e;
                }
                float inv = 1.0f / s;
                for (int i = 0; i < NCAP; ++i) cb[tid * NCAP + i] *= inv;
            }
            __syncthreads();
            for (int e = tid; e < BH * 1024; e += 256) {
                const int bl = e >> 10, col = e & 1023;
                oacc[e] = fmaf(cb[bl * NCAP + (col >> 5)], uhat[bl * UHS + col], oacc[e]);
            }
        }
        __syncthreads();
    }

    float* pp = partial + (size_t)(chunk * 2 + half) * (BH * 1024);
    for (int e = tid; e < BH * 1024; e += 256) pp[e] = oacc[e];
}

// Deterministic reduction over the 128 n-chunk partials for one batch, then
// l2_normalize (mode 0) or squash (mode 1) per capsule. grid = 32 (one WG per batch).
__global__ __launch_bounds__(256)
void caps_reduce(const float* __restrict__ partial, float* __restrict__ out, int mode)
{
    __shared__ float osum[NOUT];
    __shared__ float fac[NCAP];
    const int b = blockIdx.x;
    const int half = b >> 4, bl = b & 15;
    const int tid = threadIdx.x;

    for (int e = tid; e < NOUT; e += 256) {
        float s = 0.0f;
        const float* p = partial + (size_t)half * (BH * 1024) + (size_t)bl * 1024 + e;
        for (int c = 0; c < NCHUNK; ++c)
            s += p[(size_t)c * 2 * (BH * 1024)];
        osum[e] = s;
    }
    __syncthreads();

    if (tid < NCAP) {
        float s2 = 0.0f;
        for (int j = 0; j < CAPD; ++j) { float v = osum[tid * CAPD + j]; s2 = fmaf(v, v, s2); }
        float f;
        if (mode == 0) f = rsqrtf(fmaxf(s2, 1e-12f));                       // tf l2_normalize
        else           f = (s2 / (1.0f + s2)) * rsqrtf(s2 + 1e-7f);        // squash
        fac[tid] = f;
    }
    __syncthreads();

    for (int e = tid; e < NOUT; e += 256)
        out[(size_t)b * NOUT + e] = osum[e] * fac[e >> 5];
}

extern "C" void kernel_launch(void* const* d_in, const int* in_sizes, int n_in,
                              void* d_out, int out_size, void* d_ws, size_t ws_size,
                              hipStream_t stream)
{
    const float* U  = (const float*)d_in[0];   // u_vecs [32,2048,16]
    const float* Wt = (const float*)d_in[1];   // W      [2048,16,1024]
    float* out = (float*)d_out;                // [32,32,32]

    // workspace: partials [256][16*1024] f32 (16 MB) + o_prev [32768] f32
    float* partial = (float*)d_ws;
    float* o_prev  = partial + (size_t)(NCHUNK * 2) * (BH * 1024);

    const size_t shmem = (size_t)(BH * UHS + BH * 1024 + 2 * BH * NCAP) * sizeof(float);
    dim3 grid(NCHUNK, 2), blk(256);

    // round 0 (uniform c) -> l2 normalize
    caps_pass<<<grid, blk, shmem, stream>>>(U, Wt, nullptr, partial, 1);
    caps_reduce<<<BATCH, 256, 0, stream>>>(partial, o_prev, 0);
    // round 1 -> l2 normalize
    caps_pass<<<grid, blk, shmem, stream>>>(U, Wt, o_prev, partial, 0);
    caps_reduce<<<BATCH, 256, 0, stream>>>(partial, o_prev, 0);
    // round 2 -> squash -> d_out
    caps_pass<<<grid, blk, shmem, stream>>>(U, Wt, o_prev, partial, 0);
    caps_reduce<<<BATCH, 256, 0, stream>>>(partial, out, 1);
}